// Interaction_67525475827995
// MI455X (gfx1250) — compile-verified
//
#include <hip/hip_runtime.h>

#define N_NODES 10000
#define N_EDGES 160000
#define HID 64
#define NUM_RBF 32
#define QD 8

typedef float v2f __attribute__((ext_vector_type(2)));
typedef float v8f __attribute__((ext_vector_type(8)));
typedef unsigned int u32x4 __attribute__((ext_vector_type(4)));
typedef int i32x4 __attribute__((ext_vector_type(4)));
typedef int i32x8 __attribute__((ext_vector_type(8)));

static constexpr size_t PLANE = (size_t)N_NODES * HID;          // 640,000 floats
static constexpr size_t MSG_ELEMS = (size_t)N_NODES * 9 * HID;  // 5,760,000 floats

#if defined(__has_builtin)
#if __has_builtin(__builtin_amdgcn_tensor_load_to_lds) && \
    __has_builtin(__builtin_amdgcn_s_wait_tensorcnt)
#define HAVE_TDM 1
#endif
#endif

__device__ __forceinline__ v8f wmma4(v2f a, v2f b, v8f c) {
  return __builtin_amdgcn_wmma_f32_16x16x4_f32(false, a, false, b, (short)0, c, false, false);
}

__device__ __forceinline__ float silu_f(float x) { return x / (1.0f + __expf(-x)); }

// padded-LDS index: TDM inserts 1 DWORD of pad every 64 DWORDs
__device__ __forceinline__ int pad64(int idx) { return idx + (idx >> 6); }

// ---------------------------------------------------------------------------
// K0: zero the message accumulator
// ---------------------------------------------------------------------------
__global__ void zero_kernel(float* __restrict__ p, size_t n) {
  size_t i = (size_t)blockIdx.x * blockDim.x + threadIdx.x;
  if (i < n) p[i] = 0.0f;
}

// ---------------------------------------------------------------------------
// One MLP layer on a 16-row LDS tile, fp32 WMMA 16x16x4.
// W points at a pad64-laid-out LDS copy of a row-major [K][Nout] matrix.
// ---------------------------------------------------------------------------
__device__ __forceinline__ void mlp_layer(const float* __restrict__ in, int K,
                                          const float* __restrict__ W,
                                          const float* __restrict__ bias, int Nout,
                                          float* __restrict__ out, int lane,
                                          const float* Cmul) {
  const int m_a = lane & 15;
  const int k_off = (lane >> 4) << 1;
  const int n_d = lane & 15;
  const int m_base = (lane >> 4) << 3;
  for (int nt = 0; nt < Nout; nt += 16) {
    v8f acc = {};
    for (int kb = 0; kb < K; kb += 4) {
      const int ka = kb + k_off;
      v2f a, b;
      a.x = in[m_a * K + ka];
      a.y = in[m_a * K + ka + 1];
      b.x = W[pad64(ka * Nout + nt + n_d)];
      b.y = W[pad64((ka + 1) * Nout + nt + n_d)];
      acc = wmma4(a, b, acc);
    }
    const int colg = nt + n_d;
    const float bcol = bias[colg];
#pragma unroll
    for (int r = 0; r < 8; ++r) {
      const int m = m_base + r;
      float v = silu_f(acc[r] + bcol);
      if (Cmul) v *= Cmul[m];
      out[m * Nout + colg] = v;
    }
  }
}

// ---------------------------------------------------------------------------
// TDM: stage a [rows][cols] fp32 row-major matrix into LDS with 1-DWORD pad
// every 64 DWORDs (cols must be a multiple of 64). Caller waits on TENSORcnt.
// ---------------------------------------------------------------------------
__device__ __forceinline__ void tdm_stage_matrix(const float* gsrc, float* ldst,
                                                 int rows, int cols, int lane) {
#if defined(HAVE_TDM)
  const unsigned long long ga = (unsigned long long)(uintptr_t)gsrc;
  const unsigned int lds_addr = (unsigned int)(uintptr_t)ldst;  // low 32 = LDS offset
  u32x4 g0;
  g0[0] = 1u;                                  // count=1 (valid), no gather
  g0[1] = lds_addr;                            // lds_addr [63:32]
  g0[2] = (unsigned int)(ga & 0xFFFFFFFFull);  // global_addr low
  g0[3] = (unsigned int)((ga >> 32) & 0x1FFFFFFull) | (2u << 30);  // addr hi | type=2
  i32x8 g1;
  // data_size=4B(code2), pad_enable, pad_interval=64dw(code5), pad_amount=1dw(code0)
  g1[0] = (2 << 16) | (1 << 20) | (5 << 22);
  g1[1] = (cols & 0xFFFF) << 16;               // tensor_dim0 low16 @bit48
  g1[2] = ((cols >> 16) & 0xFFFF) | ((rows & 0xFFFF) << 16);  // dim0 hi | dim1 lo
  g1[3] = ((rows >> 16) & 0xFFFF) | ((cols & 0xFFFF) << 16);  // dim1 hi | tile_dim0
  g1[4] = rows & 0xFFFF;                       // tile_dim1 (tile_dim2=0)
  g1[5] = cols;                                // tensor_dim0_stride low32
  g1[6] = ((rows * cols) & 0xFFFF) << 16;      // stride0 hi16=0 | stride1 low16
  g1[7] = (rows * cols) >> 16;                 // stride1 upper bits
  const i32x4 z4 = {0, 0, 0, 0};
#if __clang_major__ >= 23
  const i32x8 z8 = {0, 0, 0, 0, 0, 0, 0, 0};
  __builtin_amdgcn_tensor_load_to_lds(g0, g1, z4, z4, z8, 0);
#else
  __builtin_amdgcn_tensor_load_to_lds(g0, g1, z4, z4, 0);
#endif
#else
  for (int idx = lane; idx < rows * cols; idx += 32) ldst[pad64(idx)] = gsrc[idx];
#endif
  (void)lane;
}

__device__ __forceinline__ void tdm_wait() {
#if defined(HAVE_TDM)
  __builtin_amdgcn_s_wait_tensorcnt(0);
#endif
}

// ---------------------------------------------------------------------------
// K1: normalize X, decompose into 10 planes, apply Wt0/Wt1/Wt2 via WMMA.
// Wt matrices TDM-staged into padded LDS; B-fragments come from DS.
// ---------------------------------------------------------------------------
__global__ void __launch_bounds__(32) node_prep_kernel(
    const float* __restrict__ X, const float* __restrict__ Wt0,
    const float* __restrict__ Wt1, const float* __restrict__ Wt2,
    float* __restrict__ IAS1) {
  __shared__ float stage[10][16 * HID];
  __shared__ float wts[3][64 * 64 + 64];
  const int lane = threadIdx.x;
  const int n0 = blockIdx.x * 16;

  tdm_stage_matrix(Wt0, wts[0], 64, 64, lane);
  tdm_stage_matrix(Wt1, wts[1], 64, 64, lane);
  tdm_stage_matrix(Wt2, wts[2], 64, 64, lane);

  for (int idx = lane; idx < 16 * HID; idx += 32) {
    const int r = idx >> 6, h = idx & 63;
    const size_t base = (size_t)(n0 + r) * 9 * HID + h;
    float x00 = X[base + 0 * HID], x01 = X[base + 1 * HID], x02 = X[base + 2 * HID];
    float x10 = X[base + 3 * HID], x11 = X[base + 4 * HID], x12 = X[base + 5 * HID];
    float x20 = X[base + 6 * HID], x21 = X[base + 7 * HID], x22 = X[base + 8 * HID];
    const float nrm = x00 * x00 + x01 * x01 + x02 * x02 + x10 * x10 + x11 * x11 +
                      x12 * x12 + x20 * x20 + x21 * x21 + x22 * x22;
    const float inv = 1.0f / (nrm + 1.0f);
    x00 *= inv; x01 *= inv; x02 *= inv; x10 *= inv; x11 *= inv; x12 *= inv;
    x20 *= inv; x21 *= inv; x22 *= inv;
    const float I0 = (x00 + x11 + x22) * (1.0f / 3.0f);
    stage[0][idx] = I0;
    stage[1][idx] = 0.5f * (x01 - x10);
    stage[2][idx] = 0.5f * (x02 - x20);
    stage[3][idx] = 0.5f * (x12 - x21);
    stage[4][idx] = x00 - I0;
    stage[5][idx] = 0.5f * (x01 + x10);
    stage[6][idx] = 0.5f * (x02 + x20);
    stage[7][idx] = x11 - I0;
    stage[8][idx] = 0.5f * (x12 + x21);
    stage[9][idx] = x22 - I0;
  }
  tdm_wait();
  __syncthreads();

  const int m_a = lane & 15;
  const int k_off = (lane >> 4) << 1;
  const int n_d = lane & 15;
  const int m_base = (lane >> 4) << 3;
  for (int set = 0; set < 10; ++set) {
    const float* W = (set == 0) ? wts[0] : (set < 4 ? wts[1] : wts[2]);
    float* out = IAS1 + (size_t)set * PLANE;
    for (int nt = 0; nt < HID; nt += 16) {
      v8f acc = {};
      for (int kb = 0; kb < HID; kb += 4) {
        const int ka = kb + k_off;
        v2f a, b;
        a.x = stage[set][m_a * HID + ka];
        a.y = stage[set][m_a * HID + ka + 1];
        b.x = W[pad64(ka * HID + nt + n_d)];
        b.y = W[pad64((ka + 1) * HID + nt + n_d)];
        acc = wmma4(a, b, acc);
      }
#pragma unroll
      for (int r = 0; r < 8; ++r) {
        const int m = m_base + r;
        out[(size_t)(n0 + m) * HID + nt + n_d] = acc[r];
      }
    }
  }
}

// ---------------------------------------------------------------------------
// K2: edge MLP (3 WMMA GEMM layers) + message scatter (atomic fp32 adds).
// One 8-wave block per WGP: ALL weights TDM-staged once into ~142 KB of
// shared LDS (probing the 320 KB/WG ISA limit); each wave owns a private
// 20 KB activation slice and a strided set of 16-edge tiles. No block
// barriers inside the tile loop (per-wave DS ordering suffices).
// ---------------------------------------------------------------------------
#define EDGE_BLOCKS 160
#define EDGE_WAVES 8

__global__ void __launch_bounds__(EDGE_WAVES * 32) edge_kernel(
    const float* __restrict__ charges, const float* __restrict__ edge_weight,
    const float* __restrict__ edge_attr, const int* __restrict__ edge_index,
    const float* __restrict__ Ws1, const float* __restrict__ bs1,
    const float* __restrict__ Ws2, const float* __restrict__ bs2,
    const float* __restrict__ Ws3, const float* __restrict__ bs3,
    const float* __restrict__ IAS1, float* __restrict__ msg) {
  __shared__ float w1s[48 * 64 + 48];              // 12.2 KB
  __shared__ float w2s[64 * 128 + 128];            // 32.5 KB
  __shared__ float w3s[128 * 192 + 384];           // 97.5 KB
  __shared__ float bufA[EDGE_WAVES][16 * 128];     // 8 x 8 KB
  __shared__ float bufB[EDGE_WAVES][16 * 192];     // 8 x 12 KB
  __shared__ float Cl[EDGE_WAVES][16];
  const int wave = threadIdx.x >> 5;
  const int lane = threadIdx.x & 31;
  const int* rowp = edge_index;            // destination
  const int* colp = edge_index + N_EDGES;  // source
  const float* I1 = IAS1;
  const float* A1 = IAS1 + PLANE;
  const float* S1 = IAS1 + 4 * PLANE;

  if (wave == 0) {  // wave-uniform: only wave 0 issues the TDM loads
    tdm_stage_matrix(Ws1, w1s, 48, 64, lane);
    tdm_stage_matrix(Ws2, w2s, 64, 128, lane);
    tdm_stage_matrix(Ws3, w3s, 128, 192, lane);
    tdm_wait();
  }
  __syncthreads();

  float* A_ = bufA[wave];
  float* B_ = bufB[wave];
  float* C_ = Cl[wave];

  for (int tile = blockIdx.x * EDGE_WAVES + wave; tile < N_EDGES / 16;
       tile += EDGE_BLOCKS * EDGE_WAVES) {
    const int e0 = tile * 16;

    // gather h0 = [rbf(32) | q_row(8) | q_col(8)] as [16][48]
    for (int idx = lane; idx < 16 * 48; idx += 32) {
      const int r = idx / 48, j = idx - r * 48;
      const int e = e0 + r;
      float v;
      if (j < NUM_RBF) v = edge_attr[(size_t)e * NUM_RBF + j];
      else if (j < NUM_RBF + QD) v = charges[(size_t)rowp[e] * QD + (j - NUM_RBF)];
      else v = charges[(size_t)colp[e] * QD + (j - NUM_RBF - QD)];
      A_[idx] = v;
    }
    if (lane < 16) {
      const float ew = edge_weight[e0 + lane];
      const float c = 0.5f * (__cosf(ew * 0.6283185307179586f) + 1.0f);
      C_[lane] = (ew < 5.0f) ? c : 0.0f;
    }
    mlp_layer(A_, 48, w1s, bs1, 64, B_, lane, nullptr);
    mlp_layer(B_, 64, w2s, bs2, 128, A_, lane, nullptr);
    mlp_layer(A_, 128, w3s, bs3, 192, B_, lane, C_);

    // message: w0*I*eye + w1*A + w2*S gathered at src, scattered at dst
    for (int idx = lane; idx < 16 * HID; idx += 32) {
      const int r = idx >> 6, h = idx & 63;
      const int e = e0 + r;
      const int src = colp[e], dst = rowp[e];
      const float w0 = B_[r * 192 + h];
      const float w1 = B_[r * 192 + 64 + h];
      const float w2 = B_[r * 192 + 128 + h];
      const size_t so = (size_t)src * HID + h;
      const float Iv  = w0 * I1[so];
      const float a01 = w1 * A1[so];
      const float a02 = w1 * A1[PLANE + so];
      const float a12 = w1 * A1[2 * PLANE + so];
      const float s00 = w2 * S1[so];
      const float s01 = w2 * S1[PLANE + so];
      const float s02 = w2 * S1[2 * PLANE + so];
      const float s11 = w2 * S1[3 * PLANE + so];
      const float s12 = w2 * S1[4 * PLANE + so];
      const float s22 = w2 * S1[5 * PLANE + so];
      float* md = msg + (size_t)dst * 9 * HID + h;
      unsafeAtomicAdd(md + 0 * HID, Iv + s00);
      unsafeAtomicAdd(md + 1 * HID, a01 + s01);
      unsafeAtomicAdd(md + 2 * HID, a02 + s02);
      unsafeAtomicAdd(md + 3 * HID, -a01 + s01);
      unsafeAtomicAdd(md + 4 * HID, Iv + s11);
      unsafeAtomicAdd(md + 5 * HID, a12 + s12);
      unsafeAtomicAdd(md + 6 * HID, -a02 + s02);
      unsafeAtomicAdd(md + 7 * HID, -a12 + s12);
      unsafeAtomicAdd(md + 8 * HID, Iv + s22);
    }
  }
}

// ---------------------------------------------------------------------------
// K3: Cm = msg@Y + Y@msg, normalize, decompose -> IAS2 (pre-linear)
// ---------------------------------------------------------------------------
__global__ void __launch_bounds__(256) mix_kernel(const float* __restrict__ msg,
                                                  const float* __restrict__ IAS1,
                                                  float* __restrict__ IAS2) {
  const int idx = blockIdx.x * blockDim.x + threadIdx.x;
  if (idx >= N_NODES * HID) return;
  const int node = idx >> 6, h = idx & 63;
  const size_t so = (size_t)node * HID + h;
  const float I   = IAS1[so];
  const float a01 = IAS1[PLANE + so], a02 = IAS1[2 * PLANE + so], a12 = IAS1[3 * PLANE + so];
  const float s00 = IAS1[4 * PLANE + so], s01 = IAS1[5 * PLANE + so], s02 = IAS1[6 * PLANE + so];
  const float s11 = IAS1[7 * PLANE + so], s12 = IAS1[8 * PLANE + so], s22 = IAS1[9 * PLANE + so];
  const float Y[3][3] = {{I + s00, a01 + s01, a02 + s02},
                         {-a01 + s01, I + s11, a12 + s12},
                         {-a02 + s02, -a12 + s12, I + s22}};
  float M[3][3];
  const size_t mb = (size_t)node * 9 * HID + h;
#pragma unroll
  for (int a = 0; a < 3; ++a)
#pragma unroll
    for (int b = 0; b < 3; ++b) M[a][b] = msg[mb + (a * 3 + b) * HID];
  float Cm[3][3];
  float nrm = 0.0f;
#pragma unroll
  for (int a = 0; a < 3; ++a)
#pragma unroll
    for (int b = 0; b < 3; ++b) {
      float acc = 0.0f;
#pragma unroll
      for (int c = 0; c < 3; ++c) acc += M[a][c] * Y[c][b] + Y[a][c] * M[c][b];
      Cm[a][b] = acc;
      nrm += acc * acc;
    }
  const float inv = 1.0f / (nrm + 1.0f);
  const float tr3 = (Cm[0][0] + Cm[1][1] + Cm[2][2]) * (1.0f / 3.0f);
  IAS2[so]             = tr3 * inv;
  IAS2[PLANE + so]     = 0.5f * (Cm[0][1] - Cm[1][0]) * inv;
  IAS2[2 * PLANE + so] = 0.5f * (Cm[0][2] - Cm[2][0]) * inv;
  IAS2[3 * PLANE + so] = 0.5f * (Cm[1][2] - Cm[2][1]) * inv;
  IAS2[4 * PLANE + so] = (Cm[0][0] - tr3) * inv;
  IAS2[5 * PLANE + so] = 0.5f * (Cm[0][1] + Cm[1][0]) * inv;
  IAS2[6 * PLANE + so] = 0.5f * (Cm[0][2] + Cm[2][0]) * inv;
  IAS2[7 * PLANE + so] = (Cm[1][1] - tr3) * inv;
  IAS2[8 * PLANE + so] = 0.5f * (Cm[1][2] + Cm[2][1]) * inv;
  IAS2[9 * PLANE + so] = (Cm[2][2] - tr3) * inv;
}

// ---------------------------------------------------------------------------
// K4: in-place channel mixing of IAS2 with Wt3/Wt4/Wt5 via WMMA.
// Wt matrices TDM-staged into padded LDS.
// ---------------------------------------------------------------------------
__global__ void __launch_bounds__(32) node_mix2_kernel(const float* __restrict__ Wt3,
                                                       const float* __restrict__ Wt4,
                                                       const float* __restrict__ Wt5,
                                                       float* __restrict__ IAS2) {
  __shared__ float stage[10][16 * HID];
  __shared__ float wts[3][64 * 64 + 64];
  const int lane = threadIdx.x;
  const int n0 = blockIdx.x * 16;

  tdm_stage_matrix(Wt3, wts[0], 64, 64, lane);
  tdm_stage_matrix(Wt4, wts[1], 64, 64, lane);
  tdm_stage_matrix(Wt5, wts[2], 64, 64, lane);

  for (int set = 0; set < 10; ++set)
    for (int idx = lane; idx < 16 * HID; idx += 32)
      stage[set][idx] = IAS2[(size_t)set * PLANE + (size_t)(n0 + (idx >> 6)) * HID + (idx & 63)];
  tdm_wait();
  __syncthreads();

  const int m_a = lane & 15;
  const int k_off = (lane >> 4) << 1;
  const int n_d = lane & 15;
  const int m_base = (lane >> 4) << 3;
  for (int set = 0; set < 10; ++set) {
    const float* W = (set == 0) ? wts[0] : (set < 4 ? wts[1] : wts[2]);
    float* out = IAS2 + (size_t)set * PLANE;
    for (int nt = 0; nt < HID; nt += 16) {
      v8f acc = {};
      for (int kb = 0; kb < HID; kb += 4) {
        const int ka = kb + k_off;
        v2f a, b;
        a.x = stage[set][m_a * HID + ka];
        a.y = stage[set][m_a * HID + ka + 1];
        b.x = W[pad64(ka * HID + nt + n_d)];
        b.y = W[pad64((ka + 1) * HID + nt + n_d)];
        acc = wmma4(a, b, acc);
      }
#pragma unroll
      for (int r = 0; r < 8; ++r) {
        const int m = m_base + r;
        out[(size_t)(n0 + m) * HID + nt + n_d] = acc[r];
      }
    }
  }
}

// ---------------------------------------------------------------------------
// K5: out = Xn + dX + dX@dX
// ---------------------------------------------------------------------------
__global__ void __launch_bounds__(256) out_kernel(const float* __restrict__ X,
                                                  const float* __restrict__ IAS2,
                                                  float* __restrict__ out) {
  const int idx = blockIdx.x * blockDim.x + threadIdx.x;
  if (idx >= N_NODES * HID) return;
  const int node = idx >> 6, h = idx & 63;
  const size_t base = (size_t)node * 9 * HID + h;
  float Xn[3][3];
  float nrm = 0.0f;
#pragma unroll
  for (int a = 0; a < 3; ++a)
#pragma unroll
    for (int b = 0; b < 3; ++b) {
      const float v = X[base + (a * 3 + b) * HID];
      Xn[a][b] = v;
      nrm += v * v;
    }
  const float inv = 1.0f / (nrm + 1.0f);
#pragma unroll
  for (int a = 0; a < 3; ++a)
#pragma unroll
    for (int b = 0; b < 3; ++b) Xn[a][b] *= inv;

  const size_t so = (size_t)node * HID + h;
  const float I   = IAS2[so];
  const float a01 = IAS2[PLANE + so], a02 = IAS2[2 * PLANE + so], a12 = IAS2[3 * PLANE + so];
  const float s00 = IAS2[4 * PLANE + so], s01 = IAS2[5 * PLANE + so], s02 = IAS2[6 * PLANE + so];
  const float s11 = IAS2[7 * PLANE + so], s12 = IAS2[8 * PLANE + so], s22 = IAS2[9 * PLANE + so];
  const float dX[3][3] = {{I + s00, a01 + s01, a02 + s02},
                          {-a01 + s01, I + s11, a12 + s12},
                          {-a02 + s02, -a12 + s12, I + s22}};
#pragma unroll
  for (int a = 0; a < 3; ++a)
#pragma unroll
    for (int b = 0; b < 3; ++b) {
      float acc = Xn[a][b] + dX[a][b];
#pragma unroll
      for (int c = 0; c < 3; ++c) acc += dX[a][c] * dX[c][b];
      out[base + (a * 3 + b) * HID] = acc;
    }
}

// ---------------------------------------------------------------------------
extern "C" void kernel_launch(void* const* d_in, const int* in_sizes, int n_in,
                              void* d_out, int out_size, void* d_ws, size_t ws_size,
                              hipStream_t stream) {
  (void)in_sizes; (void)n_in; (void)out_size; (void)ws_size;
  const float* X           = (const float*)d_in[0];
  const float* charges     = (const float*)d_in[1];
  const float* edge_weight = (const float*)d_in[2];
  const float* edge_attr   = (const float*)d_in[3];
  const int*   edge_index  = (const int*)d_in[4];
  const float* Ws1 = (const float*)d_in[5];
  const float* bs1 = (const float*)d_in[6];
  const float* Ws2 = (const float*)d_in[7];
  const float* bs2 = (const float*)d_in[8];
  const float* Ws3 = (const float*)d_in[9];
  const float* bs3 = (const float*)d_in[10];
  const float* Wt0 = (const float*)d_in[11];
  const float* Wt1 = (const float*)d_in[12];
  const float* Wt2 = (const float*)d_in[13];
  const float* Wt3 = (const float*)d_in[14];
  const float* Wt4 = (const float*)d_in[15];
  const float* Wt5 = (const float*)d_in[16];

  float* ws   = (float*)d_ws;
  float* msg  = ws;                 // [n][3][3][64]
  float* IAS1 = ws + MSG_ELEMS;     // 10 planes of [n][64]
  float* IAS2 = IAS1 + 10 * PLANE;  // 10 planes of [n][64]

  zero_kernel<<<(int)((MSG_ELEMS + 255) / 256), 256, 0, stream>>>(msg, MSG_ELEMS);
  node_prep_kernel<<<N_NODES / 16, 32, 0, stream>>>(X, Wt0, Wt1, Wt2, IAS1);
  edge_kernel<<<EDGE_BLOCKS, EDGE_WAVES * 32, 0, stream>>>(
      charges, edge_weight, edge_attr, edge_index,
      Ws1, bs1, Ws2, bs2, Ws3, bs3, IAS1, msg);
  mix_kernel<<<(N_NODES * HID + 255) / 256, 256, 0, stream>>>(msg, IAS1, IAS2);
  node_mix2_kernel<<<N_NODES / 16, 32, 0, stream>>>(Wt3, Wt4, Wt5, IAS2);
  out_kernel<<<(N_NODES * HID + 255) / 256, 256, 0, stream>>>(X, IAS2, (float*)d_out);
}